// SpatialTransform_51410758533745
// MI455X (gfx1250) — compile-verified
//
#include <hip/hip_runtime.h>

// Problem constants (match the reference exactly).
#define NB   16
#define H_   512
#define W_   512
#define C_   8          // channels; 8 floats = 32 bytes per pixel
#define HO_  512
#define WO_  512
#define PLANE (HO_ * WO_)          // 262144 = 2^18
#define PLANE_SHIFT 18
#define TOTAL (NB * PLANE)         // 2^22 pixels

typedef float v4f __attribute__((ext_vector_type(4)));

__device__ __forceinline__ v4f fma4(float w, v4f v, v4f acc) {
    acc.x = fmaf(w, v.x, acc.x);
    acc.y = fmaf(w, v.y, acc.y);
    acc.z = fmaf(w, v.z, acc.z);
    acc.w = fmaf(w, v.w, acc.w);
    return acc;
}

__global__ __launch_bounds__(256) void SpatialTransform_51410758533745_kernel(
    const float* __restrict__ inputs,   // (N, H, W, C) f32
    const float* __restrict__ coords,   // (N, 2, HO, WO) f32 in [-1, 1]
    float* __restrict__ out)            // (N, HO, WO, C) f32
{
    const int idx = blockIdx.x * blockDim.x + threadIdx.x;   // pixel id
    if (idx >= TOTAL) return;

    const int n = idx >> PLANE_SHIFT;          // batch
    const int p = idx & (PLANE - 1);           // pixel within (HO, WO) plane

    // ---- coords: streamed exactly once -> non-temporal loads (don't pollute L2)
    const float* cbase = coords + (size_t)n * 2 * PLANE;
    const float cy = __builtin_nontemporal_load(cbase + p);           // dim 0 -> H
    const float cx = __builtin_nontemporal_load(cbase + PLANE + p);   // dim 1 -> W

    // upscale: (c + 1) * (dim - 1) / 2
    const float fy = (cy + 1.0f) * (0.5f * (float)(H_ - 1));
    const float fx = (cx + 1.0f) * (0.5f * (float)(W_ - 1));
    const float by = floorf(fy);
    const float bx = floorf(fx);
    // Bilinear weights from UNCLIPPED corners (as in the reference):
    // off=0 -> 1-t, off=1 -> t with t = f - floor(f).
    const float ty = fy - by;
    const float tx = fx - bx;
    const float w00 = (1.0f - ty) * (1.0f - tx);
    const float w01 = (1.0f - ty) * tx;
    const float w10 = ty * (1.0f - tx);
    const float w11 = ty * tx;

    // sample() clips indices to the valid range
    int iy0 = (int)by;  iy0 = min(max(iy0, 0), H_ - 1);
    int ix0 = (int)bx;  ix0 = min(max(ix0, 0), W_ - 1);
    const int iy1 = min(iy0 + 1, H_ - 1);
    const int ix1 = min(ix0 + 1, W_ - 1);

    // ---- gathers: input footprint (128 MB) lives in the 192 MB L2 -> regular
    // temporal loads, 2x global_load_b128 per corner (32 B aligned).
    const v4f* ibase = (const v4f*)(inputs + (size_t)n * (H_ * W_ * C_));
    const int o00 = (iy0 * W_ + ix0) * 2;   // in units of v4f
    const int o01 = (iy0 * W_ + ix1) * 2;
    const int o10 = (iy1 * W_ + ix0) * 2;
    const int o11 = (iy1 * W_ + ix1) * 2;

    const v4f a0 = ibase[o00],     a1 = ibase[o00 + 1];
    const v4f b0 = ibase[o01],     b1 = ibase[o01 + 1];
    const v4f c0 = ibase[o10],     c1 = ibase[o10 + 1];
    const v4f d0 = ibase[o11],     d1 = ibase[o11 + 1];

    v4f r0 = {0.0f, 0.0f, 0.0f, 0.0f};
    v4f r1 = {0.0f, 0.0f, 0.0f, 0.0f};
    r0 = fma4(w00, a0, r0);  r1 = fma4(w00, a1, r1);
    r0 = fma4(w01, b0, r0);  r1 = fma4(w01, b1, r1);
    r0 = fma4(w10, c0, r0);  r1 = fma4(w10, c1, r1);
    r0 = fma4(w11, d0, r0);  r1 = fma4(w11, d1, r1);

    // ---- output: written exactly once -> non-temporal b128 stores so the
    // 128 MB result stream does not evict the resident input from L2.
    v4f* obase = (v4f*)out + (size_t)idx * 2;
    __builtin_nontemporal_store(r0, obase);
    __builtin_nontemporal_store(r1, obase + 1);
}

extern "C" void kernel_launch(void* const* d_in, const int* in_sizes, int n_in,
                              void* d_out, int out_size, void* d_ws, size_t ws_size,
                              hipStream_t stream) {
    (void)in_sizes; (void)n_in; (void)d_ws; (void)ws_size; (void)out_size;
    const float* inputs = (const float*)d_in[0];   // (16, 512, 512, 8) f32
    const float* coords = (const float*)d_in[1];   // (16, 2, 512, 512) f32
    float* out = (float*)d_out;                    // (16, 512, 512, 8) f32

    const int threads = 256;
    const int blocks  = TOTAL / threads;           // 2^22 / 256 = 16384
    SpatialTransform_51410758533745_kernel<<<blocks, threads, 0, stream>>>(
        inputs, coords, out);
}